// LinearObjectness_18897856103215
// MI455X (gfx1250) — compile-verified
//
#include <hip/hip_runtime.h>

// RMSNorm + GEMV fusion for:  out[b,n] = rsqrt(mean(x^2)+eps) * <x, gamma*w> + bias
// x = encoder[b, n, :], n < 2048 (encoder S-dim is 4096; only first 2048 used).
// Bandwidth-bound (128 MB @ 23.3 TB/s ~ 5.6 us); dot-product done with full-precision
// f32 WMMA (V_WMMA_F32_16X16X4_F32), sum-of-squares in VALU, gamma*w staged in LDS.

typedef float v2f __attribute__((ext_vector_type(2)));
typedef float v8f __attribute__((ext_vector_type(8)));

#define D_DIM 2048
#define S_DIM 4096
#define N_IMG 2048
#define ROWS_PER_BLOCK 16
#define WAVES 8
#define THREADS (WAVES * 32)
#define K_PER_WAVE (D_DIM / WAVES) /* 256 */
#define K_ITERS (K_PER_WAVE / 8)   /* 32: each iter covers 8 K via two 16x16x4 WMMAs */

__global__ __launch_bounds__(THREADS)
void LinearObjectness_rms_gemv_kernel(const float* __restrict__ enc,
                                      const float* __restrict__ gamma,
                                      const float* __restrict__ w,
                                      const float* __restrict__ bptr,
                                      float* __restrict__ out)
{
    __shared__ float s_gw[D_DIM];            // gamma*w, 8 KB
    __shared__ float s_dot[WAVES * 16];
    __shared__ float s_ssq[WAVES * 16];

    const int t = threadIdx.x;

    // Stage gamma*w into LDS once per block (gamma/w are L2-resident).
    for (int i = t; i < D_DIM; i += THREADS)
        s_gw[i] = gamma[i] * w[i];
    __syncthreads();

    const int wave  = t >> 5;
    const int lane  = t & 31;
    const int rowIn = lane & 15;   // M index within the 16-row tile
    const int khalf = lane >> 4;   // lane-half selects K slots {0,1} vs {2,3}

    // Global row -> encoder offset (encoder has S=4096 rows per batch, use first 2048).
    const int rowG = blockIdx.x * ROWS_PER_BLOCK + rowIn;
    const int bIdx = rowG >> 11;       // / N_IMG
    const int nIdx = rowG & (N_IMG - 1);
    const float* rowPtr = enc + (size_t)(bIdx * S_DIM + nIdx) * (size_t)D_DIM;

    // This wave's K slice; lane-half 1 takes the upper 4 of each 8-wide chunk.
    const int kstart = wave * K_PER_WAVE + khalf * 4;

    const float4* rp = (const float4*)(rowPtr + kstart);
    const float4* gp = (const float4*)(s_gw + kstart);

    v8f c0 = {0.f, 0.f, 0.f, 0.f, 0.f, 0.f, 0.f, 0.f};
    v8f c1 = {0.f, 0.f, 0.f, 0.f, 0.f, 0.f, 0.f, 0.f};
    float ssq = 0.f;

    // EXEC is all-ones here (no divergence yet) -- required for WMMA.
#pragma unroll 4
    for (int i = 0; i < K_ITERS; ++i) {
        const float4 xv = rp[i * 2];   // 8 K per iter -> stride 2 float4s
        const float4 gv = gp[i * 2];

        v2f a0 = {xv.x, xv.y};
        v2f a1 = {xv.z, xv.w};
        v2f b0 = {gv.x, gv.y};
        v2f b1 = {gv.z, gv.w};

        // D = A(16x4 f32) * B(4x16 f32) + C ; B columns all equal gamma*w,
        // so every column of D accumulates the exact f32 row dot-product.
        c0 = __builtin_amdgcn_wmma_f32_16x16x4_f32(false, a0, false, b0,
                                                   (short)0, c0, false, false);
        c1 = __builtin_amdgcn_wmma_f32_16x16x4_f32(false, a1, false, b1,
                                                   (short)0, c1, false, false);

        ssq = fmaf(xv.x, xv.x, ssq);
        ssq = fmaf(xv.y, xv.y, ssq);
        ssq = fmaf(xv.z, xv.z, ssq);
        ssq = fmaf(xv.w, xv.w, ssq);
    }

    // Fold the two lane-halves of each row's sum-of-squares.
    ssq += __shfl_xor(ssq, 16, 32);
    if (lane < 16)
        s_ssq[wave * 16 + lane] = ssq;

    // C/D layout: VGPR j, lanes 0-15 -> M=j ; lanes 16-31 -> M=j+8 (all N identical).
    v8f d = c0 + c1;
    if (lane == 0 || lane == 16) {
        float* p = s_dot + wave * 16 + (khalf << 3);
#pragma unroll
        for (int j = 0; j < 8; ++j)
            p[j] = d[j];
    }
    __syncthreads();

    // Cross-wave reduction + epilogue: 16 rows per block.
    if (t < 16) {
        float dot = 0.f, s = 0.f;
#pragma unroll
        for (int wv = 0; wv < WAVES; ++wv) {
            dot += s_dot[wv * 16 + t];
            s   += s_ssq[wv * 16 + t];
        }
        const float inv = rsqrtf(s * (1.0f / (float)D_DIM) + 1e-6f);
        out[blockIdx.x * ROWS_PER_BLOCK + t] = fmaf(dot, inv, bptr[0]);
    }
}

extern "C" void kernel_launch(void* const* d_in, const int* in_sizes, int n_in,
                              void* d_out, int out_size, void* d_ws, size_t ws_size,
                              hipStream_t stream) {
    // Inputs (setup_inputs order): encoder, objectness (unused), boxes (unused),
    // gamma, w, b(scalar).
    const float* enc   = (const float*)d_in[0];
    const float* gamma = (const float*)d_in[3];
    const float* w     = (const float*)d_in[4];
    const float* bptr  = (const float*)d_in[5];
    float* out = (float*)d_out;

    const int rows = out_size;                 // B * N_IMG = 8192
    dim3 grid(rows / ROWS_PER_BLOCK);          // 512 blocks x 16 rows
    LinearObjectness_rms_gemv_kernel<<<grid, THREADS, 0, stream>>>(enc, gamma, w, bptr, out);
    (void)in_sizes; (void)n_in; (void)d_ws; (void)ws_size;
}